// FMFMNeuronInhib_34033320854098
// MI455X (gfx1250) — compile-verified
//
#include <hip/hip_runtime.h>
#include <stdint.h>

// Problem constants (from reference setup_inputs)
#define T_STEPS  1024
#define BATCH    16384
#define TC       32                 // timesteps per staged chunk
#define NTHREADS 128                // wave32 x 4 per block
#define NCHUNK   (T_STEPS / TC)
#define ROW_FLOATS (NTHREADS * 2)   // floats per staged row (128 neurons x 2 ch)
#define ROW_BYTES  (ROW_FLOATS * 4) // 1 KB
#define CHUNK_FLOATS (TC * ROW_FLOATS)
#define CHUNK_BYTES  (CHUNK_FLOATS * 4)  // 32 KB per buffer

// Issue async global->LDS loads for one chunk (16 x b128 per thread: each
// instruction moves 16B/lane = 512B/wave; 128 threads cover 2 rows/pass).
__device__ __forceinline__ void issue_chunk_loads(
    const float* __restrict__ gbase,   // uniform: spike_seq + b0*2
    uint32_t lds_base, int chunk, int buf, int lrow, int lcol_bytes)
{
    const uint32_t grow_stride = (uint32_t)(BATCH * 2 * 4);   // 128 KB between t rows
    uint32_t goff  = (uint32_t)(chunk * TC + lrow) * grow_stride + (uint32_t)lcol_bytes;
    uint32_t laddr = lds_base + (uint32_t)buf * CHUNK_BYTES
                   + (uint32_t)lrow * ROW_BYTES + (uint32_t)lcol_bytes;
#pragma unroll
    for (int p = 0; p < TC / 2; ++p) {
        asm volatile("global_load_async_to_lds_b128 %0, %1, %2"
                     :: "v"(laddr), "v"(goff), "s"(gbase)
                     : "memory");
        goff  += 2u * grow_stride;
        laddr += 2u * ROW_BYTES;
    }
}

__global__ __launch_bounds__(NTHREADS)
void FMFMNeuronInhib_kernel(const float* __restrict__ spike_seq,
                            const float* __restrict__ w_exc,
                            const float* __restrict__ w_inh,
                            float* __restrict__ out)
{
    __shared__ float smem[2 * CHUNK_FLOATS];   // 64 KB double buffer

    const int tid = threadIdx.x;
    const int b0  = blockIdx.x * NTHREADS;
    const int b   = b0 + tid;

    // uniform weights (from device memory, per reference inputs)
    const float we0 = w_exc[0];
    const float we1 = w_exc[1];
    const float wi  = w_inh[0];

    const float* gbase = spike_seq + (size_t)b0 * 2;  // row t at +t*B*2 floats

    // lane assignment for staging: 2 rows per pass, 64 lanes x 16B per row
    const int lrow = tid >> 6;           // 0..1
    const int lcol = (tid & 63) * 16;    // byte column within 1KB row

    // LDS byte offset of smem base (low 32 bits of flat LDS address)
    const uint32_t lds_base = (uint32_t)(uintptr_t)(void*)&smem[0];

    float* __restrict__ out_spk = out;
    float* __restrict__ out_mem = out + (size_t)T_STEPS * BATCH;

    // prologue: stage chunk 0 into buffer 0
    issue_chunk_loads(gbase, lds_base, 0, 0, lrow, lcol);

    float mem = 0.0f;
    float inh = 0.0f;

    for (int c = 0; c < NCHUNK; ++c) {
        if (c + 1 < NCHUNK) {
            // stage next chunk into the other buffer, then wait only for the
            // oldest 16 async loads (this chunk's slab) to have landed in LDS
            issue_chunk_loads(gbase, lds_base, c + 1, (c + 1) & 1, lrow, lcol);
            asm volatile("s_wait_asynccnt 0x10" ::: "memory");
        } else {
            asm volatile("s_wait_asynccnt 0x0" ::: "memory");
        }
        __syncthreads();   // publish slab across all waves in the block

        const float* sb = &smem[(c & 1) * CHUNK_FLOATS];
        const int t0 = c * TC;

#pragma unroll 8
        for (int r = 0; r < TC; ++r) {
            const float2 x = *(const float2*)(&sb[r * ROW_FLOATS + tid * 2]);

            inh = 0.6f * inh + x.x;                       // BETA_INH
            const float cur   = we0 * x.x + we1 * x.y + wi * inh;
            const float reset = (mem > 1.0f) ? 1.0f : 0.0f;   // uses pre-update mem
            mem = 0.9f * mem + cur - reset;               // BETA_MEM, THRESHOLD=1
            const float spk = (mem > 1.0f) ? 1.0f : 0.0f;

            const size_t ofs = (size_t)(t0 + r) * BATCH + (size_t)b;
            __builtin_nontemporal_store(spk, out_spk + ofs);
            __builtin_nontemporal_store(mem, out_mem + ofs);
        }
        __syncthreads();   // everyone done with this buffer before it is re-staged
    }
}

extern "C" void kernel_launch(void* const* d_in, const int* in_sizes, int n_in,
                              void* d_out, int out_size, void* d_ws, size_t ws_size,
                              hipStream_t stream) {
    (void)in_sizes; (void)n_in; (void)d_ws; (void)ws_size; (void)out_size;
    const float* spike_seq = (const float*)d_in[0];
    const float* w_exc     = (const float*)d_in[1];
    const float* w_inh     = (const float*)d_in[2];
    float* out             = (float*)d_out;

    dim3 grid(BATCH / NTHREADS);   // 128 blocks
    dim3 block(NTHREADS);          // 4 wave32s
    FMFMNeuronInhib_kernel<<<grid, block, 0, stream>>>(spike_seq, w_exc, w_inh, out);
}